// Eq2to2_26817775796961
// MI455X (gfx1250) — compile-verified
//
#include <hip/hip_runtime.h>

// Problem dims (fixed by reference setup_inputs)
#define N_ 32
#define D_ 32
#define S_ 32
#define B_ 15
#define M_ 128

typedef __attribute__((ext_vector_type(2))) float v2f;
typedef __attribute__((ext_vector_type(8))) float v8f;

// ---------------------------------------------------------------------------
// Kernel 1: per-(n,d) reductions: row sums, col sums, diag, trace, total.
// One block of 128 threads per (n,d) 128x128 tile.
// ---------------------------------------------------------------------------
__global__ __launch_bounds__(128) void reduce_kernel(
    const float* __restrict__ x,
    float* __restrict__ sum_rows, float* __restrict__ sum_cols,
    float* __restrict__ diag, float* __restrict__ sum_diag,
    float* __restrict__ tot) {
  __shared__ float red[256];
  const int nd = blockIdx.x;                 // n*D + d
  const int t  = threadIdx.x;                // 0..127
  const float* xp = x + (size_t)nd * (M_ * M_);

  float csum = 0.f;                          // column t sum (coalesced pass)
  for (int i = 0; i < M_; ++i) csum += xp[i * M_ + t];
  float rsum = 0.f;                          // row t sum (tile is L2-hot)
  for (int j = 0; j < M_; ++j) rsum += xp[t * M_ + j];
  const float dg = xp[t * M_ + t];

  sum_cols[nd * M_ + t] = csum;
  sum_rows[nd * M_ + t] = rsum;
  diag[nd * M_ + t] = dg;

  red[t] = csum;            // sum over t of csum == total
  red[128 + t] = dg;        // trace
  __syncthreads();
  for (int off = 64; off > 0; off >>= 1) {
    if (t < off) { red[t] += red[t + off]; red[128 + t] += red[128 + t + off]; }
    __syncthreads();
  }
  if (t == 0) { tot[nd] = red[0]; sum_diag[nd] = red[128]; }
}

// ---------------------------------------------------------------------------
// Kernel 2: fold the 13 non-dense basis ops (+bias, diag_bias, 1/m norms)
// into rowAdd[n,s,i], colAdd[n,s,j], diagAdd[n,s,i].
// One block of 128 threads per (n,s); thread t = position along m.
// ---------------------------------------------------------------------------
__global__ __launch_bounds__(128) void combine_kernel(
    const float* __restrict__ coefs, const float* __restrict__ bias,
    const float* __restrict__ diag_bias,
    const float* __restrict__ sum_rows, const float* __restrict__ sum_cols,
    const float* __restrict__ diag, const float* __restrict__ sum_diag,
    const float* __restrict__ tot,
    float* __restrict__ rowAdd, float* __restrict__ colAdd,
    float* __restrict__ diagAdd) {
  __shared__ float cs[D_ * B_];
  const int ns = blockIdx.x;
  const int n = ns / S_, s = ns % S_;
  const int t = threadIdx.x;
  for (int e = t; e < D_ * B_; e += 128) {
    const int d = e / B_, b = e % B_;
    cs[e] = coefs[(d * S_ + s) * B_ + b];
  }
  __syncthreads();

  const float inv_m = 1.0f / (float)M_;
  const float inv_m2 = inv_m * inv_m;
  float ar = bias[s];         // row-broadcast accumulator (+ constants folded)
  float ac = 0.f;             // col-broadcast accumulator
  float ad = diag_bias[s];    // diagonal-only accumulator
  for (int d = 0; d < D_; ++d) {
    const float* cp = cs + d * B_;
    const int rd = n * D_ + d;
    const float sr = sum_rows[rd * M_ + t];
    const float sc = sum_cols[rd * M_ + t];
    const float dg = diag[rd * M_ + t];
    const float sd = sum_diag[rd] * inv_m;
    const float tt = tot[rd] * inv_m2;
    // ops 1..5 (diag_embed terms)
    ad += cp[0] * dg + cp[1] * sd + cp[2] * sr * inv_m + cp[3] * sc * inv_m + cp[4] * tt;
    // ops 6,7 (row bcast), 12 (row bcast diag), 14,15 (constants)
    ar += cp[5] * sc * inv_m + cp[6] * sr * inv_m + cp[11] * dg + cp[13] * sd + cp[14] * tt;
    // ops 8,9 (col bcast), 13 (col bcast diag)
    ac += cp[7] * sc * inv_m + cp[8] * sr * inv_m + cp[12] * dg;
  }
  rowAdd[(n * S_ + s) * M_ + t] = ar;
  colAdd[(n * S_ + s) * M_ + t] = ac;
  diagAdd[(n * S_ + s) * M_ + t] = ad;
}

// ---------------------------------------------------------------------------
// Kernel 3: dense part via fp32 WMMA + fused epilogue.
// out[n,s,i,j] = sum_d c10[d,s]*x[n,d,i,j] + c11[d,s]*x[n,d,j,i] + epilogue.
// LDS holds K-PAIR-INTERLEAVED slabs: element (k, c) lives at v2f index
// [ (k/2)*ncol + c ], component (k&1). Since every WMMA operand needs
// (K=k0, K=k0+1) with k0 even, each A/B operand is ONE aligned ds_load_b64
// straight into an even VGPR pair — no packing movs.
// One block (256 thr = 8 waves) per (n,i); per wave: one 16-wide s-tile and
// two adjacent 16-wide j-tiles; K=D=32 in 8 steps of V_WMMA_F32_16X16X4_F32.
// ---------------------------------------------------------------------------
__global__ __launch_bounds__(256) void wmma_kernel(
    const float* __restrict__ x, const float* __restrict__ coefs,
    const float* __restrict__ rowAdd, const float* __restrict__ colAdd,
    const float* __restrict__ diagAdd, float* __restrict__ out) {
  __shared__ v2f xrow2[(D_ / 2) * M_];   // (k-pair, j) : x[n,k,i,j]
  __shared__ v2f xcol2[(D_ / 2) * M_];   // (k-pair, j) : x[n,k,j,i]
  __shared__ v2f c10p[(D_ / 2) * S_];    // (k-pair, s) : coefs[k,s,9]
  __shared__ v2f c11p[(D_ / 2) * S_];    // (k-pair, s) : coefs[k,s,10]

  const int blk = blockIdx.x;
  const int n = blk / M_;
  const int i = blk % M_;
  const int t = threadIdx.x;
  const size_t xb = (size_t)n * D_ * M_ * M_;

  float* fxr = (float*)xrow2;
  float* fxc = (float*)xcol2;
  for (int e = t; e < D_ * M_; e += 256) {
    const int d = e >> 7, j = e & (M_ - 1);
    const int li = (d >> 1) * (2 * M_) + j * 2 + (d & 1);   // interleaved slot
    fxr[li] = x[xb + (size_t)d * (M_ * M_) + (size_t)i * M_ + j];   // coalesced
    fxc[li] = x[xb + (size_t)d * (M_ * M_) + (size_t)j * M_ + i];   // L2-served
  }
  float* fc0 = (float*)c10p;
  float* fc1 = (float*)c11p;
  for (int e = t; e < D_ * S_; e += 256) {
    const int d = e >> 5, s = e & (S_ - 1);
    const int li = (d >> 1) * (2 * S_) + s * 2 + (d & 1);
    fc0[li] = coefs[e * B_ + 9];
    fc1[li] = coefs[e * B_ + 10];
  }
  __syncthreads();

  const int wave = t >> 5;
  const int lane = t & 31;
  const int half = lane >> 4;            // 0: lanes 0-15, 1: lanes 16-31
  const int lj = lane & 15;
  const int s0 = (wave >> 2) * 16;       // waves 0-3 -> s rows 0..15, 4-7 -> 16..31
  const int jt0 = (wave & 3) * 2;        // two adjacent 16-wide j tiles
  const int jA = jt0 * 16 + lj;
  const int jB = (jt0 + 1) * 16 + lj;

  v8f acc0 = {};
  v8f acc1 = {};
  for (int kd = 0; kd < D_; kd += 4) {
    // This lane needs K = (kd + 2*half, kd + 2*half + 1) -> k-pair index:
    const int kp = (kd >> 1) + half;
    const v2f a  = c10p[kp * S_ + s0 + lj];
    const v2f b0 = xrow2[kp * M_ + jA];
    const v2f b1 = xrow2[kp * M_ + jB];
    acc0 = __builtin_amdgcn_wmma_f32_16x16x4_f32(false, a, false, b0, (short)0, acc0, false, false);
    acc1 = __builtin_amdgcn_wmma_f32_16x16x4_f32(false, a, false, b1, (short)0, acc1, false, false);

    const v2f a2 = c11p[kp * S_ + s0 + lj];
    const v2f b2 = xcol2[kp * M_ + jA];
    const v2f b3 = xcol2[kp * M_ + jB];
    acc0 = __builtin_amdgcn_wmma_f32_16x16x4_f32(false, a2, false, b2, (short)0, acc0, false, false);
    acc1 = __builtin_amdgcn_wmma_f32_16x16x4_f32(false, a2, false, b3, (short)0, acc1, false, false);
  }

  // Epilogue: C/D layout -> vgpr v, lanes<16: M=v, lanes>=16: M=v+8.
  for (int v = 0; v < 8; ++v) {
    const int s = s0 + v + 8 * half;
    const int base = (n * S_ + s) * M_;
    const float ra = rowAdd[base + i];
    const float da = diagAdd[base + i];
    const float oA = acc0[v] + ra + colAdd[base + jA] + ((i == jA) ? da : 0.0f);
    const float oB = acc1[v] + ra + colAdd[base + jB] + ((i == jB) ? da : 0.0f);
    out[(size_t)base * M_ + (size_t)i * M_ + jA] = oA;
    out[(size_t)base * M_ + (size_t)i * M_ + jB] = oB;
  }
}

// ---------------------------------------------------------------------------
extern "C" void kernel_launch(void* const* d_in, const int* in_sizes, int n_in,
                              void* d_out, int out_size, void* d_ws, size_t ws_size,
                              hipStream_t stream) {
  const float* x         = (const float*)d_in[0];   // (N,D,m,m)
  const float* coefs     = (const float*)d_in[1];   // (D,S,B)
  const float* bias      = (const float*)d_in[2];   // (1,S,1,1)
  const float* diag_bias = (const float*)d_in[3];   // (1,S,1,1)
  float* ws = (float*)d_ws;                         // ~3.2 MB used

  const int NDM = N_ * D_ * M_;   // 131072
  const int NSM = N_ * S_ * M_;   // 131072
  float* sum_rows = ws;
  float* sum_cols = ws + NDM;
  float* diag     = ws + 2 * NDM;
  float* sum_diag = ws + 3 * NDM;
  float* tot      = ws + 3 * NDM + N_ * D_;
  float* rowAdd   = ws + 3 * NDM + 2 * N_ * D_;
  float* colAdd   = rowAdd + NSM;
  float* diagAdd  = colAdd + NSM;

  reduce_kernel<<<N_ * D_, 128, 0, stream>>>(x, sum_rows, sum_cols, diag, sum_diag, tot);
  combine_kernel<<<N_ * S_, 128, 0, stream>>>(coefs, bias, diag_bias,
                                              sum_rows, sum_cols, diag, sum_diag, tot,
                                              rowAdd, colAdd, diagAdd);
  wmma_kernel<<<N_ * M_, 256, 0, stream>>>(x, coefs, rowAdd, colAdd, diagAdd, (float*)d_out);
}